// vq_layer_7052336300316
// MI455X (gfx1250) — compile-verified
//
#include <hip/hip_runtime.h>

// VQ-VAE vector-quantization layer for MI455X (gfx1250, wave32, WMMA).
//
// x:          [32,64,64,64] f32  -> flat [N=131072, D=64]
// embeddings: [D=64, K=512]  f32 (row-major, K contiguous)
// out:        gathered codebook columns (straight-through => values are the
//             quantized vectors), reshaped to x's shape.
//
// Strategy:
//   * argmin_k (||e_k||^2 - 2 * x.e_k); ||x||^2 is row-constant.
//   * Similarity GEMM in full f32 via v_wmma_f32_16x16x4_f32 (argmin is a
//     discrete decision; bf16 would risk flipping winners).
//   * Codebook staged ONCE per block into LDS, pre-swizzled into the exact
//     WMMA-B VGPR layout -> one conflict-free ds_load_b64 per lane per step.
//   * Two 16-row tiles per wave share every B fetch: two independent WMMA
//     accumulation chains, half the B traffic.

typedef __attribute__((ext_vector_type(2))) float v2f;
typedef __attribute__((ext_vector_type(8))) float v8f;

#define DIM      64
#define KCODES   512
#define KTILES   (KCODES / 16)     // 32
#define KSTEPS   (DIM / 4)         // 16 wmma k-steps per tile
#define WAVES    8
#define ROWS_PW  32                // rows per wave (2 x 16-row tiles)
#define ROWS_PB  (WAVES * ROWS_PW) // 256 rows per block
#define BLOCK_T  (WAVES * 32)      // 256 threads

// LDS layout (dynamic):
//   sB   : KTILES*KSTEPS*32*2 floats = 32768 f (128 KB), WMMA-B packed
//   sEn  : KCODES floats (2 KB)
//   sIdx : ROWS_PB ints (1 KB)
#define SB_FLOATS   (KTILES * KSTEPS * 32 * 2)
#define LDS_BYTES   ((SB_FLOATS + KCODES) * 4 + ROWS_PB * 4)

// ---------------------------------------------------------------------------
// Pre-pass: ||e_k||^2 for each codebook column k.
// ---------------------------------------------------------------------------
__global__ __launch_bounds__(256) void vq_enorm_kernel(
    const float* __restrict__ emb, float* __restrict__ enorm)
{
    int k = blockIdx.x * blockDim.x + threadIdx.x;
    if (k < KCODES) {
        float s = 0.0f;
#pragma unroll
        for (int d = 0; d < DIM; ++d) {
            float v = emb[(size_t)d * KCODES + k];
            s += v * v;
        }
        enorm[k] = s;
    }
}

// ---------------------------------------------------------------------------
// Main kernel.
// ---------------------------------------------------------------------------
__global__ __launch_bounds__(BLOCK_T) void vq_argmin_gather_kernel(
    const float* __restrict__ x,        // [N, 64]
    const float* __restrict__ emb,      // [64, 512]
    const float* __restrict__ enorm,    // [512]
    float*       __restrict__ out)      // [N, 64]
{
    extern __shared__ float smem[];
    float* sB  = smem;                       // WMMA-B packed codebook
    float* sEn = smem + SB_FLOATS;           // ||e||^2
    int*   sIdx = (int*)(sEn + KCODES);      // per-row argmin

    const int tid  = threadIdx.x;
    const int lane = tid & 31;
    const int wave = tid >> 5;

    // --- stage codebook into LDS, swizzled to WMMA-B layout ----------------
    // B element (k-step s within tile t): lanes 0-15 hold {k=4s,4s+1} of
    // column n = t*16+lane; lanes 16-31 hold {k=4s+2,4s+3} of n = t*16+lane-16.
    // Packed address: ((t*16 + s)*32 + lane)*2 + j  (j selects the pair half).
    for (int i = tid; i < DIM * KCODES; i += BLOCK_T) {
        const int d = i >> 9;                // row of emb (k-dim)
        const int n = i & (KCODES - 1);      // column (code)
        const float v = emb[i];              // coalesced read
        const int t  = n >> 4;
        const int bn = n & 15;
        const int s  = d >> 2;
        const int r  = d & 3;
        const int dl = bn + ((r >> 1) << 4); // dest lane
        const int j  = r & 1;
        sB[(((t * KSTEPS + s) * 32) + dl) * 2 + j] = v;
    }
    for (int i = tid; i < KCODES; i += BLOCK_T) sEn[i] = enorm[i];
    __syncthreads();

    // --- A tiles: two 16-row tiles per wave, WMMA 16x4 f32 layout ----------
    const int row_base = blockIdx.x * ROWS_PB + wave * ROWS_PW;
    const int am   = lane & 15;              // row within tile
    const int koff = (lane >> 4) << 1;       // 0 (lanes 0-15) / 2 (lanes 16-31)
    const float* xrow0 = x + (size_t)(row_base + am) * DIM;
    const float* xrow1 = xrow0 + 16 * DIM;

    v2f a0[KSTEPS], a1[KSTEPS];
#pragma unroll
    for (int s = 0; s < KSTEPS; ++s) {
        a0[s] = *(const v2f*)(xrow0 + 4 * s + koff);   // b64 loads
        a1[s] = *(const v2f*)(xrow1 + 4 * s + koff);
    }

    float minv0[8], minv1[8];
    int   mini0[8], mini1[8];
#pragma unroll
    for (int j = 0; j < 8; ++j) {
        minv0[j] = 3.4e38f; mini0[j] = 0;
        minv1[j] = 3.4e38f; mini1[j] = 0;
    }

    const int bn = lane & 15;                // this lane's column within tile

    for (int t = 0; t < KTILES; ++t) {
        const int n = t * 16 + bn;
        const float* pb = sB + (size_t)t * KSTEPS * 64;

        v8f c0 = {0.f,0.f,0.f,0.f,0.f,0.f,0.f,0.f};
        v8f c1 = {0.f,0.f,0.f,0.f,0.f,0.f,0.f,0.f};
#pragma unroll
        for (int s = 0; s < KSTEPS; ++s) {
            v2f b = *(const v2f*)(pb + (s * 32 + lane) * 2);  // ds_load_b64
            c0 = __builtin_amdgcn_wmma_f32_16x16x4_f32(
                     false, a0[s], false, b, (short)0, c0, false, false);
            c1 = __builtin_amdgcn_wmma_f32_16x16x4_f32(
                     false, a1[s], false, b, (short)0, c1, false, false);
        }

        const float en = sEn[n];
#pragma unroll
        for (int j = 0; j < 8; ++j) {
            float d0 = en - 2.0f * c0[j];
            float d1 = en - 2.0f * c1[j];
            bool l0 = d0 < minv0[j];          // strict < keeps lowest n
            bool l1 = d1 < minv1[j];
            minv0[j] = l0 ? d0 : minv0[j];
            mini0[j] = l0 ? n  : mini0[j];
            minv1[j] = l1 ? d1 : minv1[j];
            mini1[j] = l1 ? n  : mini1[j];
        }
    }

    // --- cross-lane argmin within each 16-lane half ------------------------
#pragma unroll
    for (int off = 8; off >= 1; off >>= 1) {
#pragma unroll
        for (int j = 0; j < 8; ++j) {
            float ov0 = __shfl_xor(minv0[j], off, 16);
            int   oi0 = __shfl_xor(mini0[j], off, 16);
            bool  t0  = (ov0 < minv0[j]) || (ov0 == minv0[j] && oi0 < mini0[j]);
            minv0[j] = t0 ? ov0 : minv0[j];
            mini0[j] = t0 ? oi0 : mini0[j];

            float ov1 = __shfl_xor(minv1[j], off, 16);
            int   oi1 = __shfl_xor(mini1[j], off, 16);
            bool  t1  = (ov1 < minv1[j]) || (ov1 == minv1[j] && oi1 < mini1[j]);
            minv1[j] = t1 ? ov1 : minv1[j];
            mini1[j] = t1 ? oi1 : mini1[j];
        }
    }

    if ((lane & 15) == 0) {
        const int rb = wave * ROWS_PW + ((lane >> 4) << 3);  // +0 or +8
#pragma unroll
        for (int j = 0; j < 8; ++j) {
            sIdx[rb + j]      = mini0[j];
            sIdx[rb + 16 + j] = mini1[j];
        }
    }
    __syncthreads();

    // --- gather codebook columns; coalesced 256B output writes -------------
    for (int i = tid; i < ROWS_PB * DIM; i += BLOCK_T) {
        const int r = i >> 6;                // row within block
        const int d = i & (DIM - 1);
        const int code = sIdx[r];
        out[((size_t)blockIdx.x * ROWS_PB + r) * DIM + d] =
            emb[(size_t)d * KCODES + code];
    }
}

// ---------------------------------------------------------------------------
extern "C" void kernel_launch(void* const* d_in, const int* in_sizes, int n_in,
                              void* d_out, int out_size, void* d_ws, size_t ws_size,
                              hipStream_t stream)
{
    const float* x   = (const float*)d_in[0];   // 32*64*64*64 f32
    const float* emb = (const float*)d_in[1];   // 64*512 f32
    float* out   = (float*)d_out;
    float* enorm = (float*)d_ws;                // 512 f32 scratch

    const int N = in_sizes[0] / DIM;            // 131072 rows

    vq_enorm_kernel<<<(KCODES + 255) / 256, 256, 0, stream>>>(emb, enorm);

    const int grid = N / ROWS_PB;               // 512 blocks
    vq_argmin_gather_kernel<<<grid, BLOCK_T, LDS_BYTES, stream>>>(
        x, emb, enorm, out);
}